// AttentionLiereRotator_72705206386757
// MI455X (gfx1250) — compile-verified
//
#include <hip/hip_runtime.h>
#include <hip/hip_bf16.h>

#define BATCH   4
#define SEQ     2048
#define HEADS   16
#define HEAD_DIM 128
#define BLOCK   32
#define NBLK    4
#define MPAR    496          // (32*32-32)/2
#define AXES    1024
#define SPACIAL 2
#define NVEC    (BATCH*SEQ*HEADS)   // 131072 vectors of HEAD_DIM

typedef float v2f __attribute__((ext_vector_type(2)));
typedef float v8f __attribute__((ext_vector_type(8)));

// ---------------------------------------------------------------------------
// Kernel 1: c[k,m] = sum_{a,d} skew[k,m,a,d] * a     (one block per (k,m) row)
// ---------------------------------------------------------------------------
__global__ __launch_bounds__(256) void liere_reduce_c(const float* __restrict__ skew,
                                                      float* __restrict__ c)
{
    const int km = blockIdx.x;                       // 0 .. NBLK*MPAR-1
    const float* row = skew + (size_t)km * (AXES * SPACIAL);
    const int t = threadIdx.x;
    float sum = 0.f;
    for (int a = t; a < AXES; a += 256)
        sum += (row[2 * a] + row[2 * a + 1]) * (float)a;

    __shared__ float red[256];
    red[t] = sum;
    __syncthreads();
    for (int off = 128; off > 0; off >>= 1) {
        if (t < off) red[t] += red[t + off];
        __syncthreads();
    }
    if (t == 0) c[km] = red[0];
}

// ---------------------------------------------------------------------------
// Kernel 2: build skew-symmetric S[k] and compute R[k] = expm(S[k])
// scaling-and-squaring + Taylor(12), 32x32 in LDS, one 1024-thread WG per k.
// ---------------------------------------------------------------------------
__global__ __launch_bounds__(1024) void liere_expm(const float* __restrict__ c,
                                                   float* __restrict__ Rout)
{
    const int k = blockIdx.x;
    const int j = threadIdx.x & 31;
    const int i = threadIdx.x >> 5;

    __shared__ float T [32][32];
    __shared__ float U0[32][32];
    __shared__ float U1[32][32];
    __shared__ float V0[32][32];
    __shared__ float V1[32][32];

    // skew-symmetric element (tril row-major: m = i*(i-1)/2 + j for i>j)
    float s_ij;
    if      (i > j) s_ij =  c[k * MPAR + (i * (i - 1)) / 2 + j];
    else if (i < j) s_ij = -c[k * MPAR + (j * (j - 1)) / 2 + i];
    else            s_ij = 0.f;

    // max-abs reduction (reuse U0 storage, flat)
    float* flat = &U0[0][0];
    flat[threadIdx.x] = fabsf(s_ij);
    __syncthreads();
    for (int off = 512; off > 0; off >>= 1) {
        if (threadIdx.x < (unsigned)off)
            flat[threadIdx.x] = fmaxf(flat[threadIdx.x], flat[threadIdx.x + off]);
        __syncthreads();
    }
    const float mx = flat[0];
    __syncthreads();

    int sexp = 0;
    if (mx > 0.f) {
        sexp = (int)ceilf(log2f(mx * 32.f / 0.25f));   // scale so ||T||_inf <= 0.25
        sexp = sexp < 0 ? 0 : (sexp > 48 ? 48 : sexp);
    }
    const float scale = exp2f((float)(-sexp));

    T[i][j]  = s_ij * scale;
    U0[i][j] = T[i][j];                                 // Term (t=1)
    float acc = (i == j ? 1.f : 0.f) + T[i][j];         // Acc = I + T
    __syncthreads();

    // Taylor terms t = 2..12 : Term <- (Term @ T) / t ; Acc += Term
    for (int t = 2; t <= 12; ++t) {
        float (*Tm)[32] = (t & 1) ? U1 : U0;            // current term
        float (*Tn)[32] = (t & 1) ? U0 : U1;            // next term
        float sum = 0.f;
        #pragma unroll 8
        for (int p = 0; p < 32; ++p) sum += Tm[i][p] * T[p][j];
        sum *= (1.f / (float)t);
        Tn[i][j] = sum;
        acc += sum;
        __syncthreads();
    }
    V0[i][j] = acc;
    __syncthreads();

    // repeated squaring: Acc <- Acc @ Acc, sexp times (sexp uniform in block)
    float (*A )[32] = V0;
    float (*Bq)[32] = V1;
    for (int q = 0; q < sexp; ++q) {
        float sum = 0.f;
        #pragma unroll 8
        for (int p = 0; p < 32; ++p) sum += A[i][p] * A[p][j];
        Bq[i][j] = sum;
        __syncthreads();
        float (*tmp)[32] = A; A = Bq; Bq = tmp;
    }

    Rout[(size_t)k * 1024 + i * 32 + j] = A[i][j];
}

// ---------------------------------------------------------------------------
// Kernel 3: out[v][k*32+i] = sum_j R[k][i][j] * x[v][k*32+j]
// WMMA f32 16x16x4:  D(16 vec x 16 dim) = A(16 vec x 4 j) * B(4 j x 16 dim)
// One wave handles 16 vectors, all 4 blocks. 8 waves/WG, 1024 WGs.
// ---------------------------------------------------------------------------
__global__ __launch_bounds__(256) void liere_rotate(const float* __restrict__ x,
                                                    const float* __restrict__ Rg,
                                                    float* __restrict__ out,
                                                    int nvec)
{
    __shared__ __align__(16) float Rs[NBLK][BLOCK][BLOCK];     // 16 KB
    for (int t = threadIdx.x; t < NBLK * BLOCK * BLOCK; t += 256)
        ((float*)Rs)[t] = Rg[t];
    __syncthreads();

    const int lane = threadIdx.x & 31;
    const int wave = threadIdx.x >> 5;
    const int lm   = lane & 15;       // M / N index within tile
    const int lh   = lane >> 4;       // lane-half selector
    const long vbase = ((long)blockIdx.x * 8 + wave) * 16;     // 16 vectors per wave

    if (vbase < nvec) {               // wave-uniform; EXEC stays all-ones inside
        #pragma unroll
        for (int k = 0; k < NBLK; ++k) {
            // A fragments: lane lm holds vector (vbase+lm); K pair = chunk*4 + 2*lh
            v2f afr[8];
            const float* ap = x + (vbase + lm) * HEAD_DIM + k * BLOCK + lh * 2;
            #pragma unroll
            for (int ch = 0; ch < 8; ++ch)
                afr[ch] = *(const v2f*)(ap + ch * 4);

            #pragma unroll
            for (int half = 0; half < 2; ++half) {
                const int nb = half * 16;
                v8f d = {0.f, 0.f, 0.f, 0.f, 0.f, 0.f, 0.f, 0.f};
                #pragma unroll
                for (int ch = 0; ch < 8; ++ch) {
                    // B[K][N] = R[k][nb+N][ch*4+K] ; lane lm -> N=lm, K = vgpr + 2*lh
                    v2f bfr = *(const v2f*)&Rs[k][nb + lm][ch * 4 + 2 * lh];
                    d = __builtin_amdgcn_wmma_f32_16x16x4_f32(
                            false, afr[ch], false, bfr, (short)0, d, false, false);
                }
                // D: VGPR r, lane(lh,lm) -> (vec = vbase + r + 8*lh, dim = k*32+nb+lm)
                float* obase = out + (vbase + 8 * lh) * HEAD_DIM + k * BLOCK + nb + lm;
                #pragma unroll
                for (int r = 0; r < 8; ++r)
                    obase[(long)r * HEAD_DIM] = d[r];
            }
        }
    }
}

// ---------------------------------------------------------------------------
extern "C" void kernel_launch(void* const* d_in, const int* in_sizes, int n_in,
                              void* d_out, int out_size, void* d_ws, size_t ws_size,
                              hipStream_t stream)
{
    const float* x    = (const float*)d_in[0];   // [B,SEQ,HEADS,HEAD_DIM] f32
    const float* skew = (const float*)d_in[1];   // [NBLK,MPAR,AXES,SPACIAL] f32
    float* out = (float*)d_out;

    float* ws = (float*)d_ws;
    float* c  = ws;            // NBLK*MPAR = 1984 floats
    float* R  = ws + 2048;     // NBLK*32*32 = 4096 floats

    liere_reduce_c<<<NBLK * MPAR, 256, 0, stream>>>(skew, c);
    liere_expm   <<<NBLK, 1024, 0, stream>>>(c, R);
    liere_rotate <<<(NVEC / 16) / 8, 256, 0, stream>>>(x, R, out, NVEC);
}